// TinyBird_61916248539293
// MI455X (gfx1250) — compile-verified
//
#include <hip/hip_runtime.h>

// Problem constants from the reference: B=16, C=1, H=128, W=8192, NMAX=64.
#define BB     16
#define HH     128
#define WW     8192
#define NMAX   64
#define ROWS_PER_BLOCK 2     // grid.y = H / ROWS_PER_BLOCK = 64
#define TPB    256           // 8 wave32 waves per block

// map[j] encoding: >=0 -> gather source column; -2 -> separator token; -1 -> zero
#define MAP_ZERO (-1)
#define MAP_SEP  (-2)

typedef int   v4i __attribute__((ext_vector_type(4)));
typedef float v4f __attribute__((ext_vector_type(4)));

__global__ __launch_bounds__(TPB)
void tinybird_compactify_kernel(const float* __restrict__ x,
                                const int*   __restrict__ xi,
                                const int*   __restrict__ Narr,
                                const float* __restrict__ sep_param,
                                float* __restrict__ out)
{
    __shared__ int s_xi[NMAX * 2];     // staged via async global->LDS copy
    __shared__ int s_start[NMAX];
    __shared__ int s_wid[NMAX];
    __shared__ int s_src0[NMAX];
    __shared__ int s_map[WW];          // 32 KB column map (of 320 KB WGP LDS)

    const int b   = blockIdx.x;
    const int tid = threadIdx.x;
    const int n   = Narr[b];
    const float sep = sep_param[0];

    // ---- Stage xi[b] (512 B) into LDS with the gfx1250 async copy path ----
    // Each of wave0's 32 lanes moves 16 B: global_load_async_to_lds_b128,
    // tracked by ASYNCcnt, completed with s_wait_asynccnt 0.
    if (tid < 32) {
        unsigned lds_off = (unsigned)(unsigned long long)(void*)&s_xi[0] + tid * 16u;
        unsigned voff    = tid * 16u;
        unsigned long long gbase = (unsigned long long)(xi + (size_t)b * (NMAX * 2));
        asm volatile(
            "global_load_async_to_lds_b128 %0, %1, %2\n\t"
            "s_wait_asynccnt 0"
            :
            : "v"(lds_off), "v"(voff), "s"(gbase)
            : "memory");
    }

    // ---- Init map to ZERO with 128-bit DS stores ----
    for (int j4 = tid * 4; j4 < WW; j4 += TPB * 4) {
        v4i z = {MAP_ZERO, MAP_ZERO, MAP_ZERO, MAP_ZERO};
        *(v4i*)&s_map[j4] = z;
    }
    __syncthreads();

    // ---- Sequential 64-entry prefix scan (tiny; one lane) ----
    if (tid == 0) {
        int cum = 0;
        for (int i = 0; i < n; ++i) {
            int x0 = s_xi[2 * i], x1 = s_xi[2 * i + 1];
            int w  = x1 - x0; if (w < 0) w = 0;
            s_start[i] = cum + i;          // csum - wv + idx
            s_wid[i]   = w;
            s_src0[i]  = x0;
            cum += w;
        }
    }
    __syncthreads();

    // ---- Fill map: contiguous spans per segment + separator columns ----
    for (int i = 0; i < n; ++i) {
        const int s = s_start[i], wdt = s_wid[i], src0 = s_src0[i];
        for (int t = tid; t < wdt; t += TPB) {
            const int j = s + t;
            if (j < WW) {
                int src = src0 + t;
                if (src < 0) src = 0;
                if (src > WW - 1) src = WW - 1;
                s_map[j] = src;
            }
        }
        if (tid == 0 && i < n - 1) {
            const int j = s + wdt;
            if (j < WW) s_map[j] = MAP_SEP;
        }
    }
    __syncthreads();

    // ---- xi_new tail output (concatenated flat after x_new, as float) ----
    if (blockIdx.y == 0) {
        for (int i = tid; i < NMAX; i += TPB) {
            float a = 0.f, e = 0.f;
            if (i < n) {
                a = (float)s_start[i];
                e = (float)(s_start[i] + s_wid[i]);
            }
            const size_t base = (size_t)BB * HH * WW + ((size_t)b * NMAX + i) * 2;
            out[base]     = a;
            out[base + 1] = e;
        }
    }

    // ---- Bandwidth-bound gather/stream: float4 per thread per iter ----
    // Branchless: clamp index to >=0, load unconditionally (always in-bounds),
    // then v_cndmask-select sep/zero. NT hints: every byte of x is read once,
    // every byte of out written once -> don't pollute L2 (128 MB vs 192 MB).
    const int h0 = blockIdx.y * ROWS_PER_BLOCK;
    for (int r = 0; r < ROWS_PER_BLOCK; ++r) {
        const size_t rowoff = ((size_t)b * HH + (h0 + r)) * (size_t)WW;
        const float* __restrict__ xrow = x + rowoff;
        float* __restrict__ orow       = out + rowoff;
        for (int j4 = tid * 4; j4 < WW; j4 += TPB * 4) {
            const v4i m = *(const v4i*)&s_map[j4];   // ds_load_b128
            const int ix = (m.x > 0) ? m.x : 0;
            const int iy = (m.y > 0) ? m.y : 0;
            const int iz = (m.z > 0) ? m.z : 0;
            const int iw = (m.w > 0) ? m.w : 0;
            const float gx = __builtin_nontemporal_load(&xrow[ix]);
            const float gy = __builtin_nontemporal_load(&xrow[iy]);
            const float gz = __builtin_nontemporal_load(&xrow[iz]);
            const float gw = __builtin_nontemporal_load(&xrow[iw]);
            v4f v;
            v.x = (m.x >= 0) ? gx : ((m.x == MAP_SEP) ? sep : 0.f);
            v.y = (m.y >= 0) ? gy : ((m.y == MAP_SEP) ? sep : 0.f);
            v.z = (m.z >= 0) ? gz : ((m.z == MAP_SEP) ? sep : 0.f);
            v.w = (m.w >= 0) ? gw : ((m.w == MAP_SEP) ? sep : 0.f);
            __builtin_nontemporal_store(v, (v4f*)&orow[j4]);   // b128 th:NT
        }
    }
}

extern "C" void kernel_launch(void* const* d_in, const int* in_sizes, int n_in,
                              void* d_out, int out_size, void* d_ws, size_t ws_size,
                              hipStream_t stream) {
    (void)in_sizes; (void)n_in; (void)out_size; (void)d_ws; (void)ws_size;
    const float* x   = (const float*)d_in[0];
    const int*   xi  = (const int*)  d_in[1];
    const int*   N   = (const int*)  d_in[2];
    const float* sep = (const float*)d_in[3];
    float* out = (float*)d_out;

    dim3 grid(BB, HH / ROWS_PER_BLOCK);   // 16 x 64 = 1024 blocks
    dim3 block(TPB);                      // 8 wave32 waves
    tinybird_compactify_kernel<<<grid, block, 0, stream>>>(x, xi, N, sep, out);
}